// BSPLoss_80814104641648
// MI455X (gfx1250) — compile-verified
//
#include <hip/hip_runtime.h>
#include <hip/hip_bf16.h>

typedef __attribute__((ext_vector_type(2))) float v2f;
typedef __attribute__((ext_vector_type(8))) float v8f;

#define PST 80      // LDS panel row stride in floats; 80 % 32 == 16 -> conflict-free frag reads
#define TILE 64     // block tile (64x64 of C), 16x16 tile grid over D=1024
#define GD 1024     // Gram matrix dimension D
#define NTILE 16    // GD / TILE
#define TRI 136     // NTILE*(NTILE+1)/2 upper-triangular tiles
#define KSPLIT 4    // K-split for the big SYRK (occupancy)

// CDNA5 async global->LDS path (ASYNCcnt), guarded so host pass / older
// toolchains fall back to the synchronous staging path.
#if defined(__AMDGCN__) && __has_builtin(__builtin_amdgcn_global_load_async_to_lds_b128) && __has_builtin(__builtin_amdgcn_s_wait_asynccnt)
#define HAVE_ASYNC 1
typedef __attribute__((ext_vector_type(4))) int v4i;
#else
#define HAVE_ASYNC 0
#endif

// ---------------------------------------------------------------------------
// Fragment loader: CDNA5 fp32 A/B 16x4 (4x16) layout (ISA 7.12.2):
//   lane L holds column (L&15); VGPR v holds K = ((L>>4)*2 + v)
// ---------------------------------------------------------------------------
__device__ __forceinline__ v2f lds_frag(const float* p, int k0, int c0) {
  const int lane = threadIdx.x & 31;
  const int kk = k0 + ((lane >> 4) << 1);
  const int c  = c0 + (lane & 15);
  v2f r;
  r.x = p[kk * PST + c];
  r.y = p[(kk + 1) * PST + c];
  return r;
}

__device__ __forceinline__ void compute_kblock(const float* Ap, const float* Bp,
                                               int wm, int wn, v8f (&acc)[2][2]) {
#pragma unroll
  for (int k0 = 0; k0 < 32; k0 += 4) {
    v2f a0 = lds_frag(Ap, k0, wm);
    v2f a1 = lds_frag(Ap, k0, wm + 16);
    v2f b0 = lds_frag(Bp, k0, wn);
    v2f b1 = lds_frag(Bp, k0, wn + 16);
    acc[0][0] = __builtin_amdgcn_wmma_f32_16x16x4_f32(false, a0, false, b0, (short)0, acc[0][0], false, false);
    acc[0][1] = __builtin_amdgcn_wmma_f32_16x16x4_f32(false, a0, false, b1, (short)0, acc[0][1], false, false);
    acc[1][0] = __builtin_amdgcn_wmma_f32_16x16x4_f32(false, a1, false, b0, (short)0, acc[1][0], false, false);
    acc[1][1] = __builtin_amdgcn_wmma_f32_16x16x4_f32(false, a1, false, b1, (short)0, acc[1][1], false, false);
  }
}

#if HAVE_ASYNC
// Issue the 32x64 A/B panels for k-block `kb` into LDS buffers (async, ASYNCcnt).
__device__ __forceinline__ void issue_panels_async(const float* A, int kb, int bm0, int bn0,
                                                   float* Ap, float* Bp) {
  const int t = threadIdx.x;
#pragma unroll
  for (int j = 0; j < 4; ++j) {
    const int i   = t + j * 128;          // 0..511 float4 slots
    const int row = i >> 4;               // 16 float4 per 64-float row
    const int c4  = (i & 15) << 2;
    const float* ga = A + (size_t)(kb + row) * GD + bm0 + c4;
    const float* gb = A + (size_t)(kb + row) * GD + bn0 + c4;
    __builtin_amdgcn_global_load_async_to_lds_b128(
        (v4i*)ga, (v4i*)&Ap[row * PST + c4], 0, 0);
    __builtin_amdgcn_global_load_async_to_lds_b128(
        (v4i*)gb, (v4i*)&Bp[row * PST + c4], 0, 0);
  }
}
#endif

__device__ __forceinline__ void load_panels_sync(const float* A, int kb, int bm0, int bn0,
                                                 float* Ap, float* Bp) {
  const int t = threadIdx.x;
#pragma unroll
  for (int j = 0; j < 4; ++j) {
    const int i   = t + j * 128;
    const int row = i >> 4;
    const int c4  = (i & 15) << 2;
    const float* ga = A + (size_t)(kb + row) * GD + bm0 + c4;
    const float* gb = A + (size_t)(kb + row) * GD + bn0 + c4;
    *(float4*)(&Ap[row * PST + c4]) = *(const float4*)ga;
    *(float4*)(&Bp[row * PST + c4]) = *(const float4*)gb;
  }
}

// ---------------------------------------------------------------------------
// Symmetric C = scale * (A^T A): only upper-triangular 64x64 tiles are
// computed (blockIdx.x in [0,136)), results mirrored to the lower triangle.
// K is processed in gridDim.y chunks of kChunk rows; chunk y writes C_y.
// Works for G = F^T F (K=8192, 4 chunks) and H*H = H^T H (K=1024, 1 chunk).
// ---------------------------------------------------------------------------
__global__ __launch_bounds__(128) void gram_sym_kernel(
    const float* __restrict__ A,
    float* __restrict__ C0, float* __restrict__ C1,
    float* __restrict__ C2, float* __restrict__ C3,
    int kChunk, const float* __restrict__ scaleptr)
{
  float* C = (blockIdx.y == 0) ? C0 : (blockIdx.y == 1) ? C1
           : (blockIdx.y == 2) ? C2 : C3;
  const int kStart = blockIdx.y * kChunk;
  const int kEnd   = kStart + kChunk;

  // triangle index -> (ti, tj), ti <= tj
  int tt = blockIdx.x, ti = 0;
  while (tt >= (NTILE - ti)) { tt -= (NTILE - ti); ++ti; }
  const int tj  = ti + tt;
  const int bm0 = ti * TILE;
  const int bn0 = tj * TILE;

  const int t    = threadIdx.x;         // 0..127 (4 wave32s)
  const int wave = t >> 5;
  const int wm   = (wave >> 1) * 32;
  const int wn   = (wave & 1) * 32;

  __shared__ float Ap[2][32 * PST];
  __shared__ float Bp[2][32 * PST];

  v8f acc[2][2] = {};

#if HAVE_ASYNC
  // Double-buffered pipeline: async global->LDS for k-block i+1 overlaps
  // the WMMA mainloop on k-block i. One barrier per k-block.
  issue_panels_async(A, kStart, bm0, bn0, Ap[0], Bp[0]);
  int buf = 0;
  for (int kb = kStart; kb < kEnd; kb += 32) {
    __builtin_amdgcn_s_wait_asynccnt(0);
    __syncthreads();
    if (kb + 32 < kEnd)
      issue_panels_async(A, kb + 32, bm0, bn0, Ap[buf ^ 1], Bp[buf ^ 1]);
    compute_kblock(Ap[buf], Bp[buf], wm, wn, acc);
    buf ^= 1;
  }
#else
  for (int kb = kStart; kb < kEnd; kb += 32) {
    __syncthreads();
    load_panels_sync(A, kb, bm0, bn0, Ap[0], Bp[0]);
    __syncthreads();
    compute_kblock(Ap[0], Bp[0], wm, wn, acc);
  }
#endif

  const float s2 = scaleptr ? scaleptr[0] : 1.0f;
  const int lane = t & 31;
  const int colL = lane & 15;
  const int rowH = (lane >> 4) << 3;    // C/D layout: VGPR r -> M = r (+8 hi lanes)
#pragma unroll
  for (int i = 0; i < 2; ++i)
#pragma unroll
    for (int j = 0; j < 2; ++j) {
      const int rbase = bm0 + wm + i * 16 + rowH;
      const int cbase = bn0 + wn + j * 16 + colL;
#pragma unroll
      for (int r = 0; r < 8; ++r) {
        const float val = acc[i][j][r] * s2;
        C[(size_t)(rbase + r) * GD + cbase] = val;
        if (ti != tj)
          C[(size_t)cbase * GD + (rbase + r)] = val;   // mirror to lower tri
      }
    }
}

// G += (p1 + p2 + p3): combine K-split partial Gram matrices
__global__ __launch_bounds__(256) void add4_kernel(
    float* __restrict__ g, const float* __restrict__ a,
    const float* __restrict__ b, const float* __restrict__ c, int n)
{
  for (int i = blockIdx.x * 256 + threadIdx.x; i < n; i += gridDim.x * 256)
    g[i] = g[i] + a[i] + b[i] + c[i];
}

// ---------------- deterministic two-stage reductions ------------------------
__global__ __launch_bounds__(256) void sumsq_partial_kernel(
    const float* __restrict__ x, int n, float* __restrict__ partials)
{
  __shared__ float red[256];
  float s = 0.0f;
  for (int i = blockIdx.x * 256 + threadIdx.x; i < n; i += gridDim.x * 256)
    s += x[i] * x[i];
  red[threadIdx.x] = s;
  __syncthreads();
  for (int off = 128; off > 0; off >>= 1) {
    if (threadIdx.x < off) red[threadIdx.x] += red[threadIdx.x + off];
    __syncthreads();
  }
  if (threadIdx.x == 0) partials[blockIdx.x] = red[0];
}

__global__ __launch_bounds__(256) void finalize_inv_kernel(
    const float* __restrict__ partials, int n, float* __restrict__ out)
{
  __shared__ float red[256];
  float s = 0.0f;
  for (int i = threadIdx.x; i < n; i += 256) s += partials[i];
  red[threadIdx.x] = s;
  __syncthreads();
  for (int off = 128; off > 0; off >>= 1) {
    if (threadIdx.x < off) red[threadIdx.x] += red[threadIdx.x + off];
    __syncthreads();
  }
  if (threadIdx.x == 0) out[0] = 1.0f / fmaxf(red[0], 1e-30f);  // 1/||H||_F^2
}

// y = M * x (1024x1024 row-major); useOnes -> x treated as all-ones
__global__ __launch_bounds__(128) void matvec_kernel(
    const float* __restrict__ M, const float* __restrict__ x,
    float* __restrict__ y, int useOnes)
{
  __shared__ float red[128];
  const int r = blockIdx.x;
  const float* row = M + (size_t)r * GD;
  float s = 0.0f;
  for (int c = threadIdx.x; c < GD; c += 128)
    s += row[c] * (useOnes ? 1.0f : x[c]);
  red[threadIdx.x] = s;
  __syncthreads();
  for (int off = 64; off > 0; off >>= 1) {
    if (threadIdx.x < off) red[threadIdx.x] += red[threadIdx.x + off];
    __syncthreads();
  }
  if (threadIdx.x == 0) y[r] = red[0];
}

// accum += weight * (v.u)/(v.v)   (Rayleigh quotient vs original G, u = G v)
__global__ __launch_bounds__(256) void rayleigh_acc_kernel(
    const float* __restrict__ v, const float* __restrict__ u,
    float weight, float* __restrict__ accum)
{
  __shared__ float r1[256];
  __shared__ float r2[256];
  float s1 = 0.0f, s2 = 0.0f;
  for (int i = threadIdx.x; i < GD; i += 256) {
    const float vi = v[i];
    s1 += vi * u[i];
    s2 += vi * vi;
  }
  r1[threadIdx.x] = s1;
  r2[threadIdx.x] = s2;
  __syncthreads();
  for (int off = 128; off > 0; off >>= 1) {
    if (threadIdx.x < off) {
      r1[threadIdx.x] += r1[threadIdx.x + off];
      r2[threadIdx.x] += r2[threadIdx.x + off];
    }
    __syncthreads();
  }
  if (threadIdx.x == 0)
    accum[0] += weight * (r1[0] / fmaxf(r2[0], 1e-30f));
}

__global__ void zero_kernel(float* p) { p[0] = 0.0f; }
__global__ void copyout_kernel(const float* a, float* out) { out[0] = a[0]; }

// ---------------------------------------------------------------------------
extern "C" void kernel_launch(void* const* d_in, const int* in_sizes, int n_in,
                              void* d_out, int out_size, void* d_ws, size_t ws_size,
                              hipStream_t stream) {
  (void)in_sizes; (void)n_in; (void)out_size; (void)ws_size;

  float* ws   = (float*)d_ws;
  float* G    = ws;                      // 1M floats (also K-split partial 0)
  float* Ha   = ws + (1u << 20);         // 1M floats (K-split partial 1, then ping)
  float* Hb   = ws + (2u << 20);         // 1M floats (K-split partial 2, then pong)
  float* Gx   = ws + (3u << 20);         // 1M floats (K-split partial 3)
  float* vec1 = ws + (4u << 20);         // 1024
  float* vec2 = vec1 + 1024;             // 1024
  float* vec3 = vec2 + 1024;             // 1024
  float* partials = vec3 + 1024;         // 256
  float* scale    = partials + 256;      // 1
  float* accum    = scale + 1;           // 1

  zero_kernel<<<1, 1, 0, stream>>>(accum);

  const float weights[3] = {1.0f, 0.5f, 0.5f};

  for (int m = 0; m < 3; ++m) {
    const float* F = (const float*)d_in[m];

    // G = F^T F via fp32 WMMA: triangular tiles x 4-way K-split, then combine.
    gram_sym_kernel<<<dim3(TRI, KSPLIT), 128, 0, stream>>>(
        F, G, Ha, Hb, Gx, 8192 / KSPLIT, nullptr);
    add4_kernel<<<256, 256, 0, stream>>>(G, Ha, Hb, Gx, GD * GD);

    // 10 rounds of normalized squaring: H <- (H/||H||_F)^2  => P ~ G^1024
    const float* src = G;
    float* dst = Ha;
    float* other = Hb;
    for (int r = 0; r < 10; ++r) {
      sumsq_partial_kernel<<<256, 256, 0, stream>>>(src, GD * GD, partials);
      finalize_inv_kernel<<<1, 256, 0, stream>>>(partials, 256, scale);
      gram_sym_kernel<<<dim3(TRI, 1), 128, 0, stream>>>(
          src, dst, dst, dst, dst, GD, scale);
      src = dst;
      float* tmp = dst; dst = other; other = tmp;
    }
    const float* P = src;  // normalized G^1024

    // v ~ top eigenvector: v = P*(P*ones)  (direction of G^2048 * ones)
    matvec_kernel<<<GD, 128, 0, stream>>>(P, nullptr, vec1, 1);
    matvec_kernel<<<GD, 128, 0, stream>>>(P, vec1, vec2, 0);
    // Rayleigh quotient against the ORIGINAL G for full fp32 accuracy
    matvec_kernel<<<GD, 128, 0, stream>>>(G, vec2, vec3, 0);
    rayleigh_acc_kernel<<<1, 256, 0, stream>>>(vec2, vec3, weights[m], accum);
  }

  copyout_kernel<<<1, 1, 0, stream>>>(accum, (float*)d_out);
}